// ManifoldHyperConnections_40372692582485
// MI455X (gfx1250) — compile-verified
//
#include <hip/hip_runtime.h>

typedef __attribute__((ext_vector_type(2))) float v2f;
typedef __attribute__((ext_vector_type(8))) float v8f;

#define NS 4            // N streams
#define MS 2            // M
#define IN_DIM 4096
#define BLK_D 1024
#define SK_ITERS 20
#define EPSF 1.1920929e-07f
#define ROWS 16         // batch rows per block in proj kernel
#define NCOLS 32        // 8 (read_in) + 8 (write_out) + 16 (stream_mixing)

// ---------------------------------------------------------------------------
// Kernel A: fused RMS sum-of-squares + (16 x 4096) @ (4096 x 32) projection via
// V_WMMA_F32_16X16X4_F32, then per-row sigmoid/Sinkhorn and 4x4 combine matrix
// C = wo @ ri + sinkhorn(sm), written to workspace (16 floats per batch row).
// ---------------------------------------------------------------------------
__global__ __launch_bounds__(256) void mhc_proj_kernel(
    const float* __restrict__ x,
    const float* __restrict__ read_in,      const float* __restrict__ alpha_ri,
    const float* __restrict__ write_out,    const float* __restrict__ alpha_wo,
    const float* __restrict__ stream_mix,   const float* __restrict__ alpha_sm,
    const float* __restrict__ W_ri,
    const float* __restrict__ W_wo,
    const float* __restrict__ W_sm,
    float* __restrict__ Cmat)               // (Btot, 16)
{
    __shared__ float hpart[8][ROWS][NCOLS]; // per-wave partial H   (16 KB)
    __shared__ float sqpart[8][32];         // per-lane x^2 partial (1 KB)
    __shared__ float Hf[ROWS][NCOLS];       // reduced H            (2 KB)

    const int tid      = threadIdx.x;
    const int wave     = tid >> 5;
    const int lane     = tid & 31;
    const int mrow     = lane & 15;          // matrix row (0..15) this lane serves
    const int koff     = (lane >> 4) << 1;   // lanes 0-15 -> K 0,1 ; lanes 16-31 -> K 2,3
    const int row_base = blockIdx.x * ROWS;

    // ---- A-fragment row pointer (x) and B-fragment column pointers (weights)
    const float* arow = x + (size_t)(row_base + mrow) * IN_DIM;
    // column tile 0 -> cols 0..15 : 0..7 = W_read_in rows, 8..15 = W_write_out rows
    const float* bp0 = (mrow < 8) ? (W_ri + (size_t)mrow * IN_DIM)
                                  : (W_wo + (size_t)(mrow - 8) * IN_DIM);
    // column tile 1 -> cols 16..31 : W_stream_mixing rows 0..15
    const float* bp1 = W_sm + (size_t)mrow * IN_DIM;

    v8f acc0 = {0.f,0.f,0.f,0.f,0.f,0.f,0.f,0.f};
    v8f acc1 = {0.f,0.f,0.f,0.f,0.f,0.f,0.f,0.f};
    float sq = 0.f;

    const int kbeg = wave * (IN_DIM / 8);    // 512-wide K slice per wave
    const int kend = kbeg + (IN_DIM / 8);
    for (int k = kbeg; k < kend; k += 4) {
        v2f a  = *(const v2f*)(arow + k + koff);
        v2f b0 = *(const v2f*)(bp0  + k + koff);
        v2f b1 = *(const v2f*)(bp1  + k + koff);
        sq += a.x * a.x + a.y * a.y;         // RMS sum-of-squares, fused (x read once)
        acc0 = __builtin_amdgcn_wmma_f32_16x16x4_f32(
                   false, a, false, b0, (short)0, acc0, false, false);
        acc1 = __builtin_amdgcn_wmma_f32_16x16x4_f32(
                   false, a, false, b1, (short)0, acc1, false, false);
    }

    // ---- spill per-wave results to LDS (C/D layout: lanes<16 M=r, lanes>=16 M=r+8)
    {
        const int mhi = (lane >> 4) << 3;
        const int nc  = lane & 15;
        #pragma unroll
        for (int r = 0; r < 8; ++r) {
            hpart[wave][mhi + r][nc]      = acc0[r];
            hpart[wave][mhi + r][16 + nc] = acc1[r];
        }
        sqpart[wave][lane] = sq;
    }
    __syncthreads();

    // ---- deterministic cross-wave reduction of H (512 entries / 256 threads)
    for (int e = tid; e < ROWS * NCOLS; e += 256) {
        const int m = e >> 5, n = e & 31;
        float s = 0.f;
        #pragma unroll
        for (int w = 0; w < 8; ++w) s += hpart[w][m][n];
        Hf[m][n] = s;
    }
    __syncthreads();

    // ---- per-row small math: scale, sigmoids, Sinkhorn, combine into C (4x4)
    if (tid < ROWS) {
        const int row = tid;
        float ssq = 0.f;
        #pragma unroll
        for (int w = 0; w < 8; ++w) ssq += sqpart[w][row] + sqpart[w][row + 16];
        const float scale = rsqrtf(ssq * (1.0f / IN_DIM) + EPSF);

        const float a_ri = alpha_ri[0], a_wo = alpha_wo[0], a_sm = alpha_sm[0];

        float ri[MS][NS], wo[NS][MS], p[NS][NS];
        #pragma unroll
        for (int c = 0; c < 8; ++c) {
            const int n = c >> 1, m = c & 1;
            const float hri = a_ri * (Hf[row][c] * scale) + read_in[c];
            ri[m][n] = 1.0f / (1.0f + __expf(-hri));           // sigmoid, transposed
            const float hwo = a_wo * (Hf[row][8 + c] * scale) + write_out[c];
            wo[n][m] = 2.0f / (1.0f + __expf(-hwo));           // 2*sigmoid
        }
        #pragma unroll
        for (int c = 0; c < 16; ++c) {
            const int i = c >> 2, j = c & 3;
            p[i][j] = __expf(a_sm * (Hf[row][16 + c] * scale) + stream_mix[c]);
        }
        for (int it = 0; it < SK_ITERS; ++it) {
            #pragma unroll
            for (int i = 0; i < NS; ++i) {
                const float inv = 1.0f / (p[i][0] + p[i][1] + p[i][2] + p[i][3]);
                p[i][0] *= inv; p[i][1] *= inv; p[i][2] *= inv; p[i][3] *= inv;
            }
            #pragma unroll
            for (int j = 0; j < NS; ++j) {
                const float inv = 1.0f / (p[0][j] + p[1][j] + p[2][j] + p[3][j]);
                p[0][j] *= inv; p[1][j] *= inv; p[2][j] *= inv; p[3][j] *= inv;
            }
        }
        float* Cb = Cmat + ((size_t)row_base + row) * 16;
        #pragma unroll
        for (int i = 0; i < NS; ++i)
            #pragma unroll
            for (int j = 0; j < NS; ++j) {
                float c = p[i][j];
                #pragma unroll
                for (int m = 0; m < MS; ++m) c += wo[i][m] * ri[m][j];
                Cb[i * 4 + j] = c;
            }
    }
}

// ---------------------------------------------------------------------------
// Kernel B: out[b] = C[b] (4x4) @ x3[b] (4x1024). Pure streaming, coalesced.
// ---------------------------------------------------------------------------
__global__ __launch_bounds__(256) void mhc_apply_kernel(
    const float* __restrict__ x,
    const float* __restrict__ Cmat,
    float* __restrict__ out)
{
    __shared__ float C[16];
    const int tid  = threadIdx.x;
    const size_t b = blockIdx.x;
    if (tid < 16) C[tid] = Cmat[b * 16 + tid];
    __syncthreads();

    const float* xb = x   + b * (size_t)IN_DIM;
    float*       ob = out + b * (size_t)IN_DIM;

    #pragma unroll
    for (int it = 0; it < BLK_D / 256; ++it) {
        const int d = tid + it * 256;
        const float x0 = xb[d];
        const float x1 = xb[BLK_D + d];
        const float x2 = xb[2 * BLK_D + d];
        const float x3v = xb[3 * BLK_D + d];
        #pragma unroll
        for (int i = 0; i < NS; ++i) {
            ob[i * BLK_D + d] = C[i*4+0]*x0 + C[i*4+1]*x1 + C[i*4+2]*x2 + C[i*4+3]*x3v;
        }
    }
}

extern "C" void kernel_launch(void* const* d_in, const int* in_sizes, int n_in,
                              void* d_out, int out_size, void* d_ws, size_t ws_size,
                              hipStream_t stream) {
    const float* x          = (const float*)d_in[0];
    const float* read_in    = (const float*)d_in[1];
    const float* alpha_ri   = (const float*)d_in[2];
    const float* write_out  = (const float*)d_in[3];
    const float* alpha_wo   = (const float*)d_in[4];
    const float* stream_mix = (const float*)d_in[5];
    const float* alpha_sm   = (const float*)d_in[6];
    const float* W_ri       = (const float*)d_in[7];
    const float* W_wo       = (const float*)d_in[8];
    const float* W_sm       = (const float*)d_in[9];
    float* out  = (float*)d_out;
    float* Cws  = (float*)d_ws;          // Btot * 16 floats = 1 MB

    const int Btot  = in_sizes[0] / IN_DIM;   // 16384
    const int tiles = Btot / ROWS;            // 1024

    mhc_proj_kernel<<<tiles, 256, 0, stream>>>(
        x, read_in, alpha_ri, write_out, alpha_wo, stream_mix, alpha_sm,
        W_ri, W_wo, W_sm, Cws);
    mhc_apply_kernel<<<Btot, 256, 0, stream>>>(x, Cws, out);
}